// SubjectLayers_63428077027510
// MI455X (gfx1250) — compile-verified
//
#include <hip/hip_runtime.h>

typedef __attribute__((ext_vector_type(2))) float v2f;
typedef __attribute__((ext_vector_type(4))) float v4f;
typedef __attribute__((ext_vector_type(8))) float v8f;

#define B_    64
#define C_    256
#define T_    2048
#define D_    256

#define BLK_D 128
#define BLK_T 128
#define KC    16                   // K-chunk (channels) staged per LDS tile
#define K2    (KC / 2)             // K pairs per tile
#define NKT   (C_ / KC)            // 16 K-tiles
// interleaved row: [k2][col][2]; stride 2*128+32 floats -> 32-bank offset between
// consecutive k2 rows => the two half-waves of a b64 fragment load hit disjoint banks
#define LDSW  (2 * BLK_T + 32)

__global__ __launch_bounds__(256) void subject_layers_wmma_f32(
    const float* __restrict__ x,        // [B, C, T]
    const int*   __restrict__ subjects, // [B]
    const float* __restrict__ weights,  // [N, C, D]
    const float* __restrict__ bias,     // [N, D]
    float* __restrict__ out)            // [B, D, T]
{
    __shared__ __align__(16) float Xs[2][K2][LDSW];  // x tile,  K-pair interleaved
    __shared__ __align__(16) float Ws[2][K2][LDSW];  // w tile,  K-pair interleaved
    __shared__ float Bs[BLK_D];

    const int tid  = threadIdx.x;
    const int lane = tid & 31;
    const int wave = tid >> 5;

    const int b  = blockIdx.z;
    const int d0 = blockIdx.y * BLK_D;
    const int t0 = blockIdx.x * BLK_T;

    const int subj = subjects[b];
    const float* __restrict__ wp = weights + (size_t)subj * (C_ * D_);
    const float* __restrict__ xp = x + (size_t)b * ((size_t)C_ * T_);

    // wave sub-tile: 64 rows (D) x 32 cols (T) of the 128x128 block tile
    const int wd = (wave & 1) * 64;
    const int wt = (wave >> 1) * 32;

    const int kq = lane >> 4;   // half-wave: K pair {kk,kk+1} vs {kk+2,kk+3}
    const int ml = lane & 15;   // M (A) / N (B,C,D) index within fragment

    v8f acc[4][2];
#pragma unroll
    for (int m = 0; m < 4; ++m)
#pragma unroll
        for (int n = 0; n < 2; ++n)
            acc[m][n] = (v8f)(0.0f);

    // staging registers: one C-row pair of 4 columns per array per thread
    v4f xlo, xhi, wlo, whi;
    const int k2u  = tid >> 5;        // 0..7  : K-pair row this thread stages
    const int col4 = tid & 31;        // 0..31 : float4 column

    auto gload = [&](int kt) {
        const int c = kt * KC + 2 * k2u;
        xlo = *(const v4f*)(xp + (size_t)c * T_ + t0 + col4 * 4);
        xhi = *(const v4f*)(xp + (size_t)(c + 1) * T_ + t0 + col4 * 4);
        wlo = *(const v4f*)(wp + (size_t)c * D_ + d0 + col4 * 4);
        whi = *(const v4f*)(wp + (size_t)(c + 1) * D_ + d0 + col4 * 4);
    };
    auto sstore = [&](int buf) {
        // interleave the two K rows: in LDS each column holds row K in slot 0 and row K+1 in slot 1
        float* px = &Xs[buf][k2u][col4 * 8];
        float* pw = &Ws[buf][k2u][col4 * 8];
        v4f xi0 = {xlo.x, xhi.x, xlo.y, xhi.y};
        v4f xi1 = {xlo.z, xhi.z, xlo.w, xhi.w};
        v4f wi0 = {wlo.x, whi.x, wlo.y, whi.y};
        v4f wi1 = {wlo.z, whi.z, wlo.w, whi.w};
        ((v4f*)px)[0] = xi0;
        ((v4f*)px)[1] = xi1;
        ((v4f*)pw)[0] = wi0;
        ((v4f*)pw)[1] = wi1;
    };

    if (tid < BLK_D) Bs[tid] = bias[subj * D_ + d0 + tid];

    gload(0);
    sstore(0);
    __syncthreads();

    for (int kt = 0; kt < NKT; ++kt) {
        const int cur = kt & 1;
        if (kt + 1 < NKT) gload(kt + 1);   // overlap next global fetch with WMMA

#pragma unroll
        for (int kk = 0; kk < KC; kk += 4) {
            const int r2 = (kk >> 1) + kq;           // k2 row for this half-wave
            v2f a[4], bf[2];
#pragma unroll
            for (int m = 0; m < 4; ++m)
                a[m] = *(const v2f*)&Ws[cur][r2][2 * (wd + 16 * m + ml)];
#pragma unroll
            for (int n = 0; n < 2; ++n)
                bf[n] = *(const v2f*)&Xs[cur][r2][2 * (wt + 16 * n + ml)];
#pragma unroll
            for (int m = 0; m < 4; ++m)
#pragma unroll
                for (int n = 0; n < 2; ++n)
                    acc[m][n] = __builtin_amdgcn_wmma_f32_16x16x4_f32(
                        false, a[m], false, bf[n], (short)0, acc[m][n],
                        false, false);
        }

        __syncthreads();                     // all waves done reading buffer we overwrite next
        if (kt + 1 < NKT) sstore((kt + 1) & 1);
        __syncthreads();                     // stores visible before next compute
    }

    // Epilogue: C/D layout -> VGPR r holds M = r (lanes 0-15) / r+8 (lanes 16-31), N = lane%16
#pragma unroll
    for (int m = 0; m < 4; ++m) {
#pragma unroll
        for (int n = 0; n < 2; ++n) {
            const int t = t0 + wt + 16 * n + ml;
#pragma unroll
            for (int r = 0; r < 8; ++r) {
                const int dl = wd + 16 * m + r + 8 * kq;   // d - d0
                out[(size_t)b * ((size_t)D_ * T_) + (size_t)(d0 + dl) * T_ + t] =
                    acc[m][n][r] + Bs[dl];
            }
        }
    }
}

extern "C" void kernel_launch(void* const* d_in, const int* in_sizes, int n_in,
                              void* d_out, int out_size, void* d_ws, size_t ws_size,
                              hipStream_t stream) {
    const float* x        = (const float*)d_in[0];
    const int*   subjects = (const int*)d_in[1];
    const float* weights  = (const float*)d_in[2];
    const float* bias     = (const float*)d_in[3];
    float*       out      = (float*)d_out;

    dim3 grid(T_ / BLK_T, D_ / BLK_D, B_);   // (16, 2, 64)
    dim3 block(256);
    subject_layers_wmma_f32<<<grid, block, 0, stream>>>(x, subjects, weights, bias, out);
}